// LearnableGMM_59957743452263
// MI455X (gfx1250) — compile-verified
//
#include <hip/hip_runtime.h>
#include <hip/hip_bf16.h>
#include <math.h>

typedef __attribute__((ext_vector_type(2))) float v2f;
typedef __attribute__((ext_vector_type(8))) float v8f;

#define KC 8   // components
#define DD 16  // dimensions

// ---------------------------------------------------------------------------
// Setup: cov = S S^T, Cholesky L, A = L^{-1}, bias = -A*mu, per-comp constant
// One block, 256 threads. Tiny (~100K FLOPs).
// ---------------------------------------------------------------------------
__global__ __launch_bounds__(256) void gmm_setup_kernel(
    const float* __restrict__ logits,   // [8]
    const float* __restrict__ means,    // [8,16]
    const float* __restrict__ scales,   // [8,16,16]
    float* __restrict__ Aout,           // [8,16,16] row-major
    float* __restrict__ bneg,           // [8,16]  = -(A_k mu_k)
    float* __restrict__ ck)             // [8]
{
    __shared__ float Ls[KC][DD][DD];   // cov, then Cholesky in place
    __shared__ float As[KC][DD][DD];   // A = L^{-1}
    __shared__ float hld[KC];          // half logdet
    const int t = threadIdx.x;

    // 1) covariance: cov[k][i][j] = sum_d S[k][i][d]*S[k][j][d]  (2048 entries)
    for (int e = t; e < KC * DD * DD; e += blockDim.x) {
        const int k = e >> 8, i = (e >> 4) & 15, j = e & 15;
        const float* si = scales + (k * DD + i) * DD;
        const float* sj = scales + (k * DD + j) * DD;
        float acc = 0.f;
        #pragma unroll
        for (int d = 0; d < DD; ++d) acc += si[d] * sj[d];
        Ls[k][i][j] = acc;
    }
    __syncthreads();

    // 2) Cholesky, one component per thread (k < 8)
    if (t < KC) {
        const int k = t;
        float hl = 0.f;
        for (int j = 0; j < DD; ++j) {
            float djj = Ls[k][j][j];
            for (int p = 0; p < j; ++p) djj -= Ls[k][j][p] * Ls[k][j][p];
            const float ljj = sqrtf(djj);
            Ls[k][j][j] = ljj;
            hl += logf(ljj);
            const float inv = 1.f / ljj;
            for (int i = j + 1; i < DD; ++i) {
                float v = Ls[k][i][j];
                for (int p = 0; p < j; ++p) v -= Ls[k][i][p] * Ls[k][j][p];
                Ls[k][i][j] = v * inv;
            }
        }
        hld[k] = hl;
    }
    __syncthreads();

    // 3) A = L^{-1}: forward-substitute unit columns; one (k,col) per thread
    if (t < KC * DD) {
        const int k = t >> 4, c = t & 15;
        for (int i = 0; i < DD; ++i) {
            float v;
            if (i < c) {
                v = 0.f;
            } else {
                v = (i == c) ? 1.f : 0.f;
                for (int j = c; j < i; ++j) v -= Ls[k][i][j] * As[k][j][c];
                v /= Ls[k][i][i];
            }
            As[k][i][c] = v;
        }
    }
    __syncthreads();

    // 4) publish A, bias, constants
    for (int e = t; e < KC * DD * DD; e += blockDim.x) {
        const int k = e >> 8, i = (e >> 4) & 15, j = e & 15;
        Aout[e] = As[k][i][j];
    }
    if (t < KC * DD) {
        const int k = t >> 4, i = t & 15;
        float acc = 0.f;
        #pragma unroll
        for (int j = 0; j < DD; ++j) acc += As[k][i][j] * means[k * DD + j];
        bneg[k * DD + i] = -acc;
    }
    if (t < KC) {
        float mx = logits[0];
        for (int k = 1; k < KC; ++k) mx = fmaxf(mx, logits[k]);
        float s = 0.f;
        for (int k = 0; k < KC; ++k) s += expf(logits[k] - mx);
        const float lse = mx + logf(s);
        // c_k = log-softmax(logits)_k - half_logdet_k - (D/2)*log(2*pi)
        ck[t] = (logits[t] - lse) - hld[t] - 8.0f * logf(6.28318530717958647692f);
    }
}

// ---------------------------------------------------------------------------
// Main: per 16-point tile, per component: D = A_k * X^T + (-A_k mu_k)
// via 4x V_WMMA_F32_16X16X4_F32; maha = column sum of squares; logsumexp.
// One wave (32 lanes) per tile; A/bias fragments hoisted across tiles.
// ---------------------------------------------------------------------------
__global__ __launch_bounds__(256) void gmm_logprob_kernel(
    const float* __restrict__ data,   // [N,16]
    const float* __restrict__ A,      // [8,16,16]
    const float* __restrict__ bneg,   // [8,16]
    const float* __restrict__ ck,     // [8]
    float* __restrict__ out,          // [N]
    int npoints)
{
    const int lane = threadIdx.x & 31;
    const int hh   = lane >> 4;        // lane half: selects K in {0,1} vs {2,3}
    const int m    = lane & 15;        // A row index == point column index
    const int wavesPerBlock = blockDim.x >> 5;
    const int gwave  = blockIdx.x * wavesPerBlock + (threadIdx.x >> 5);
    const int nwaves = gridDim.x * wavesPerBlock;
    const int ntiles = (npoints + 15) >> 4;

    // A fragments (WMMA A layout: lane m holds row M=m; vgpr r = K = r + 2*hh,
    // so the needed pair per k-step s is the consecutive (4s+2hh, 4s+2hh+1)).
    v2f afrag[KC][4];
    #pragma unroll
    for (int k = 0; k < KC; ++k)
        #pragma unroll
        for (int s = 0; s < 4; ++s)
            afrag[k][s] = *(const v2f*)(A + ((k * DD + m) << 4) + 4 * s + 2 * hh);

    // C-accumulator init = bias: C[r] = -(A_k mu_k)[r + 8*hh]  (C/D layout)
    v8f cinit[KC];
    #pragma unroll
    for (int k = 0; k < KC; ++k) {
        const float* bp = bneg + k * DD + 8 * hh;
        #pragma unroll
        for (int r = 0; r < 8; ++r) cinit[k][r] = bp[r];
    }
    float ckv[KC];
    #pragma unroll
    for (int k = 0; k < KC; ++k) ckv[k] = ck[k];

    for (int tile = gwave; tile < ntiles; tile += nwaves) {
        int p = tile * 16 + m;
        int pc = p < npoints ? p : (npoints - 1);   // clamp loads, EXEC untouched
        const float* xp = data + pc * DD;           // this lane's point row

        // B fragments: B[d][n] = X[n][d]; same (r + 2*hh) striping as A.
        v2f bfrag[4];
        #pragma unroll
        for (int s = 0; s < 4; ++s)
            bfrag[s] = *(const v2f*)(xp + 4 * s + 2 * hh);

        float lp[KC];
        #pragma unroll
        for (int k = 0; k < KC; ++k) {
            v8f c = cinit[k];
            #pragma unroll
            for (int s = 0; s < 4; ++s)
                c = __builtin_amdgcn_wmma_f32_16x16x4_f32(
                        /*neg_a=*/false, afrag[k][s],
                        /*neg_b=*/false, bfrag[s],
                        /*c_mod=*/(short)0, c,
                        /*reuse_a=*/false, /*reuse_b=*/false);
            float ss = 0.f;
            #pragma unroll
            for (int r = 0; r < 8; ++r) ss += c[r] * c[r];
            ss += __shfl_xor(ss, 16, 32);   // combine row halves (M=r vs r+8)
            lp[k] = ckv[k] - 0.5f * ss;
        }

        // logsumexp over the 8 components
        float mx = lp[0];
        #pragma unroll
        for (int k = 1; k < KC; ++k) mx = fmaxf(mx, lp[k]);
        float s = 0.f;
        #pragma unroll
        for (int k = 0; k < KC; ++k) s += expf(lp[k] - mx);
        const float res = mx + logf(s);

        if (hh == 0 && p < npoints) out[p] = res;
    }
}

// ---------------------------------------------------------------------------
extern "C" void kernel_launch(void* const* d_in, const int* in_sizes, int n_in,
                              void* d_out, int out_size, void* d_ws, size_t ws_size,
                              hipStream_t stream) {
    const float* data   = (const float*)d_in[0];   // [N,16]
    const float* logits = (const float*)d_in[1];   // [8]
    const float* means  = (const float*)d_in[2];   // [8,16]
    const float* scales = (const float*)d_in[3];   // [8,16,16]
    float* out = (float*)d_out;
    const int npoints = in_sizes[0] / DD;

    float* ws   = (float*)d_ws;
    float* A    = ws;              // 2048 floats
    float* bneg = ws + 2048;       // 128 floats
    float* ck   = ws + 2048 + 128; // 8 floats

    gmm_setup_kernel<<<1, 256, 0, stream>>>(logits, means, scales, A, bneg, ck);

    const int blocks = 512;  // 4096 waves -> ~8 tiles per wave, amortizes A frags
    gmm_logprob_kernel<<<blocks, 256, 0, stream>>>(data, A, bneg, ck, out, npoints);
}